// niw_reg_85968065397105
// MI455X (gfx1250) — compile-verified
//
#include <hip/hip_runtime.h>
#include <math.h>

// ---------------------------------------------------------------------------
// NIW evidential loss, rewritten as four global sums over B*N = 1M elements.
//   S0 = sum( psi_diag - log(c) )          c = lmbda*(nu - n - 1)
//   S1 = sum( c * exp(-psi_diag) )         ( = sum 1/niw_var )
//   S2 = sum( d*d * c * exp(-psi_diag) )   d = mu - nan_to_num(y)
//   S3 = sum( |d| * (lmbda + nu) )
// temporal = 0.5*(S0 - Bn*log(s1) - Bn + s1*S1 + S2);  error = S3/B
// Memory-bound: ~16MB contiguous + ~1M scattered 4B diag reads (~64-128MB of
// cachelines) => ~5us at 23.3 TB/s. WMMA f32 16x16x4 used as the wave-level
// reduction engine (A = accumulators, B = column-select ones matrix).
// ---------------------------------------------------------------------------

typedef __attribute__((ext_vector_type(2))) float v2f;
typedef __attribute__((ext_vector_type(4))) float v4f;
typedef __attribute__((ext_vector_type(8))) float v8f;

#define NROWS     8192
#define NCOL      128
#define BN_TOTAL  (NROWS * NCOL)   // 1,048,576 elements
#define BLK       256              // 8 waves per block (wave32)
#define NB1       1024             // kernel-1 grid: 262144 threads = BN/4 groups
#define SIGMA1_F  0.01f

// Wave32 reduction of four per-lane accumulators using two WMMA passes.
// A (16x4 f32): lane m<16 holds {A[m,0],A[m,1]}, lane m+16 holds {A[m,2],A[m,3]}.
// B (4x16): ones at (0,0),(1,0) -> column0 sums a+b of lane m ; (2,1),(3,1) -> column1
// sums a+b of lane m+16. D[m,0] = a(m)+b(m), D[m,1] = a(m+16)+b(m+16). Sum D's 8 VGPRs
// in-lane, fold lane<->lane+16, then lane n holds the sum of column n.
__device__ __forceinline__ void wave_reduce4_wmma(float a0, float a1, float a2, float a3,
                                                  float& r0, float& r1, float& r2, float& r3) {
  const int lane = threadIdx.x & 31;
  const float sel = (lane == 0 || lane == 17) ? 1.0f : 0.0f;
  v2f bsel; bsel[0] = sel; bsel[1] = sel;
  v2f A01;  A01[0]  = a0;  A01[1]  = a1;
  v2f A23;  A23[0]  = a2;  A23[1]  = a3;
  v8f cz = {0.f, 0.f, 0.f, 0.f, 0.f, 0.f, 0.f, 0.f};
  v8f d1 = __builtin_amdgcn_wmma_f32_16x16x4_f32(false, A01, false, bsel,
                                                 (short)0, cz, false, false);
  v8f d2 = __builtin_amdgcn_wmma_f32_16x16x4_f32(false, A23, false, bsel,
                                                 (short)0, cz, false, false);
  float t1 = d1[0] + d1[1] + d1[2] + d1[3] + d1[4] + d1[5] + d1[6] + d1[7];
  float t2 = d2[0] + d2[1] + d2[2] + d2[3] + d2[4] + d2[5] + d2[6] + d2[7];
  t1 += __shfl_xor(t1, 16, 32);
  t2 += __shfl_xor(t2, 16, 32);
  r0 = __shfl(t1, 0, 32);   // column 0 -> sum of a0+a1 over lanes 0-15
  r1 = __shfl(t1, 1, 32);   // column 1 -> sum of a0+a1 over lanes 16-31
  r2 = __shfl(t2, 0, 32);   // sum of a2+a3 over lanes 0-15
  r3 = __shfl(t2, 1, 32);   // sum of a2+a3 over lanes 16-31
}

__global__ __launch_bounds__(BLK) void niw_partial_kernel(
    const float* __restrict__ y, const float* __restrict__ mu,
    const float* __restrict__ lmbda, const float* __restrict__ psi,
    const float* __restrict__ nu, v4f* __restrict__ partials) {
  const int tid = blockIdx.x * BLK + threadIdx.x;
  const int nthreads = NB1 * BLK;                 // 262144 == BN_TOTAL/4
  float a0 = 0.f, a1 = 0.f, a2 = 0.f, a3 = 0.f;

  const v4f* y4 = reinterpret_cast<const v4f*>(y);
  const v4f* m4 = reinterpret_cast<const v4f*>(mu);
  const v4f* l4 = reinterpret_cast<const v4f*>(lmbda);
  const v4f* n4 = reinterpret_cast<const v4f*>(nu);

  // Uniform trip count across all threads -> EXEC stays all-ones for WMMA.
  for (int g = tid; g < BN_TOTAL / 4; g += nthreads) {
    v4f yv = __builtin_nontemporal_load(&y4[g]);
    v4f mv = __builtin_nontemporal_load(&m4[g]);
    v4f lv = __builtin_nontemporal_load(&l4[g]);
    v4f nv = __builtin_nontemporal_load(&n4[g]);
    const int e0 = g * 4;
#pragma unroll
    for (int k = 0; k < 4; ++k) {
      const int e = e0 + k;
      // psi diag: psi[b,j,j] at b*N*N + j*(N+1) == e*N + (e & (N-1))
      const size_t off = (size_t)e * NCOL + (size_t)(e & (NCOL - 1));
      const float pd = __builtin_nontemporal_load(psi + off);
      float yy = yv[k];
      yy = (yy == yy) ? yy : 0.0f;                // nan_to_num(y)
      const float dd  = mv[k] - yy;
      const float c   = lv[k] * (nv[k] - (float)(NCOL + 1));
      const float inv = c * expf(-pd);            // 1 / niw_var
      a0 += pd - logf(c);                         // log(niw_var), algebraically
      a1 += inv;
      a2 += dd * dd * inv;
      a3 += fabsf(dd) * (lv[k] + nv[k]);
    }
  }

  float r0, r1, r2, r3;
  wave_reduce4_wmma(a0, a1, a2, a3, r0, r1, r2, r3);

  __shared__ v4f wpart[BLK / 32];
  const int w = threadIdx.x >> 5;
  if ((threadIdx.x & 31) == 0) {
    v4f t; t[0] = r0; t[1] = r1; t[2] = r2; t[3] = r3;
    wpart[w] = t;
  }
  __syncthreads();
  if (threadIdx.x == 0) {
    v4f s = wpart[0];
#pragma unroll
    for (int i = 1; i < BLK / 32; ++i) {
      v4f q = wpart[i];
      s[0] += q[0]; s[1] += q[1]; s[2] += q[2]; s[3] += q[3];
    }
    partials[blockIdx.x] = s;
  }
}

__global__ __launch_bounds__(BLK) void niw_final_kernel(
    const v4f* __restrict__ partials, float* __restrict__ out) {
  float a0 = 0.f, a1 = 0.f, a2 = 0.f, a3 = 0.f;
  for (int i = threadIdx.x; i < NB1; i += BLK) {  // uniform: 4 iters each
    v4f p = partials[i];
    a0 += p[0]; a1 += p[1]; a2 += p[2]; a3 += p[3];
  }
  float r0, r1, r2, r3;
  wave_reduce4_wmma(a0, a1, a2, a3, r0, r1, r2, r3);

  __shared__ v4f wpart[BLK / 32];
  const int w = threadIdx.x >> 5;
  if ((threadIdx.x & 31) == 0) {
    v4f t; t[0] = r0; t[1] = r1; t[2] = r2; t[3] = r3;
    wpart[w] = t;
  }
  __syncthreads();
  if (threadIdx.x == 0) {
    double S0 = 0.0, S1 = 0.0, S2 = 0.0, S3 = 0.0;
#pragma unroll
    for (int i = 0; i < BLK / 32; ++i) {
      v4f q = wpart[i];
      S0 += (double)q[0]; S1 += (double)q[1];
      S2 += (double)q[2]; S3 += (double)q[3];
    }
    const double Bn = (double)BN_TOTAL;
    const double LOG_SIGMA1 = -4.605170185988091;   // log(0.01)
    const double temporal = 0.5 * (S0 - Bn * LOG_SIGMA1 - Bn
                                   + (double)SIGMA1_F * S1 + S2);
    const double error = S3 / (double)NROWS;        // mean over batch
    const double loss = 0.5 * temporal + 0.5 * error;
    out[0] = (float)loss;
    out[1] = (float)temporal;
    out[2] = (float)error;
  }
}

extern "C" void kernel_launch(void* const* d_in, const int* in_sizes, int n_in,
                              void* d_out, int out_size, void* d_ws, size_t ws_size,
                              hipStream_t stream) {
  (void)in_sizes; (void)n_in; (void)out_size; (void)ws_size;
  const float* y     = (const float*)d_in[0];
  const float* mu    = (const float*)d_in[1];
  const float* lmbda = (const float*)d_in[2];
  const float* psi   = (const float*)d_in[3];
  const float* nu    = (const float*)d_in[4];
  float* out = (float*)d_out;
  v4f* partials = (v4f*)d_ws;                       // NB1 * 16 bytes = 16 KB

  niw_partial_kernel<<<NB1, BLK, 0, stream>>>(y, mu, lmbda, psi, nu, partials);
  niw_final_kernel<<<1, BLK, 0, stream>>>(partials, out);
}